// pix_attn_27822798143556
// MI455X (gfx1250) — compile-verified
//
#include <hip/hip_runtime.h>

// CDNA5 / gfx1250, wave32. fp32 WMMA path: v_wmma_f32_16x16x4_f32.
typedef float v2f __attribute__((ext_vector_type(2)));
typedef float v8f __attribute__((ext_vector_type(8)));

#define LDA 66   // padded row stride for 64-wide LDS tiles (bank-conflict avoidance, 8B aligned)
#define LDQ 130  // padded row stride for 128-wide LDS tiles

// A-fragment (16x4 f32): lane m = lane&15; VGPR0 = K0|K2, VGPR1 = K1|K3 across half-waves.
__device__ __forceinline__ v2f a_frag(const float* __restrict__ A, int lane, int kk) {
  const int m  = lane & 15;
  const int k0 = (kk << 2) + ((lane >> 4) << 1);
  const float* p = A + m * LDA + k0;
  v2f a; a.x = p[0]; a.y = p[1];
  return a;
}

// B-fragment (4x16 f32) from row-major W[k][n], leading dim ldw.
__device__ __forceinline__ v2f b_frag(const float* __restrict__ W, int ldw, int col, int lane, int kk) {
  const int n  = col + (lane & 15);
  const int k0 = (kk << 2) + ((lane >> 4) << 1);
  v2f b; b.x = W[k0 * ldw + n]; b.y = W[(k0 + 1) * ldw + n];
  return b;
}

__global__ __launch_bounds__(256)
void pix_attn_fused_kernel(const float* __restrict__ x,
                           const float* __restrict__ Wqk, const float* __restrict__ bqk,
                           const float* __restrict__ Wkv, const float* __restrict__ bkv,
                           const float* __restrict__ Wpr, const float* __restrict__ bpr,
                           float* __restrict__ out) {
  __shared__ float s_xf[10 * 16 * LDA];   // x_f tokens: [token 0..9][pixel 0..15][C], token0 = pooled mean
  __shared__ float s_xraw[16 * LDA];      // x_raw
  __shared__ float s_qq[16 * LDQ];        // qk projection (q | q_)
  __shared__ float s_kv[10 * 16 * LDQ];   // kv projection (k | v)
  __shared__ float s_ao[16 * LDA];        // attention output (pre-proj)

  const int tid  = threadIdx.x;
  const int lane = tid & 31;
  const int wv   = tid >> 5;

  // 16 consecutive pixels along W: shared (b, h), w = w0 + m.
  const int p0  = blockIdx.x << 4;
  const int bi  = p0 >> 12;
  const int rem = p0 & 4095;
  const int h   = rem >> 6;
  const int w0  = rem & 63;

  const float* __restrict__ xb = x + (size_t)bi * (64 * 64 * 64);

  // ---- gather x_raw (coalesced along w) ----
  for (int i = tid; i < 16 * 64; i += 256) {
    const int c = i >> 4, m = i & 15;
    s_xraw[m * LDA + c] = xb[(c * 64 + h) * 64 + w0 + m];
  }
  // ---- gather 3x3 reflect-padded windows with the torch-faithful (C,9)->(9,C) scramble ----
  for (int j = tid; j < 64 * 9 * 16; j += 256) {
    const int c0 = j / 144;             // source channel
    const int r2 = j - c0 * 144;
    const int l0 = r2 >> 4;             // window position 0..8
    const int m  = r2 & 15;             // pixel in tile (fastest -> coalesced reads)
    const int di = l0 / 3 - 1, dj = l0 % 3 - 1;
    int sh = h + di;      sh = sh < 0 ? 1 : (sh > 63 ? 62 : sh);   // reflect pad=1
    int sw = w0 + m + dj; sw = sw < 0 ? 1 : (sw > 63 ? 62 : sw);
    const float val = xb[(c0 * 64 + sh) * 64 + sw];
    const int f = c0 * 9 + l0;          // flat index in (C,9) order
    const int l = f >> 6, c = f & 63;   // reinterpreted as (9, C)
    s_xf[((l + 1) * 16 + m) * LDA + c] = val;
  }
  __syncthreads();
  // ---- pooled/global token = mean over 9 window tokens ----
  for (int i = tid; i < 16 * 64; i += 256) {
    const int m = i >> 6, c = i & 63;
    float s = 0.f;
#pragma unroll
    for (int l = 1; l <= 9; ++l) s += s_xf[(l * 16 + m) * LDA + c];
    s_xf[m * LDA + c] = s * (1.0f / 9.0f);
  }
  __syncthreads();

  // ---- qk projection: (16x64) @ (64x128); wave wv owns cols [16*wv, 16*wv+16) ----
  {
    const int n0 = wv << 4;
    v2f bf[16];
#pragma unroll
    for (int kk = 0; kk < 16; ++kk) bf[kk] = b_frag(Wqk, 128, n0, lane, kk);
    v8f acc = {};
#pragma unroll
    for (int kk = 0; kk < 16; ++kk)
      acc = __builtin_amdgcn_wmma_f32_16x16x4_f32(
          false, a_frag(s_xraw, lane, kk), false, bf[kk], (short)0, acc, false, false);
    const float bias = bqk[n0 + (lane & 15)];
#pragma unroll
    for (int g = 0; g < 8; ++g) {
      const int m = g + ((lane >> 4) << 3);
      s_qq[m * LDQ + n0 + (lane & 15)] = acc[g] + bias;
    }
  }

  // ---- kv projection: 10 tokens x (16x64)@(64x128); B-fragments reused across tokens ----
  {
    const int n0 = wv << 4;
    v2f bf[16];
#pragma unroll
    for (int kk = 0; kk < 16; ++kk) bf[kk] = b_frag(Wkv, 128, n0, lane, kk);
    const float bias = bkv[n0 + (lane & 15)];
    for (int l = 0; l < 10; ++l) {
      const float* __restrict__ A = s_xf + l * 16 * LDA;
      v8f acc = {};
#pragma unroll
      for (int kk = 0; kk < 16; ++kk)
        acc = __builtin_amdgcn_wmma_f32_16x16x4_f32(
            false, a_frag(A, lane, kk), false, bf[kk], (short)0, acc, false, false);
#pragma unroll
      for (int g = 0; g < 8; ++g) {
        const int m = g + ((lane >> 4) << 3);
        s_kv[(l * 16 + m) * LDQ + n0 + (lane & 15)] = acc[g] + bias;
      }
    }
  }
  __syncthreads();

  // ---- attention: one thread per (pixel, head); 11-way softmax (10 tokens + q_value) ----
  if (tid < 128) {
    const int m = tid >> 3, hh = tid & 7;
    const float scale = 0.35355339059327373f;  // 8^-0.5
    float q[8], qp[8];
#pragma unroll
    for (int d = 0; d < 8; ++d) {
      q[d]  = s_qq[m * LDQ + hh * 8 + d];
      qp[d] = s_qq[m * LDQ + 64 + hh * 8 + d];
    }
    float lg[11];
    float qv = 0.f;
#pragma unroll
    for (int d = 0; d < 8; ++d) qv += q[d] * qp[d];
#pragma unroll
    for (int n = 0; n < 10; ++n) {
      float s = 0.f;
#pragma unroll
      for (int d = 0; d < 8; ++d) s += q[d] * s_kv[(n * 16 + m) * LDQ + hh * 8 + d];
      lg[n] = s * scale;
    }
    lg[10] = qv;
    float mx = lg[0];
#pragma unroll
    for (int n = 1; n < 11; ++n) mx = fmaxf(mx, lg[n]);
    float den = 0.f;
#pragma unroll
    for (int n = 0; n < 11; ++n) { lg[n] = __expf(lg[n] - mx); den += lg[n]; }
    const float inv = 1.f / den;
#pragma unroll
    for (int d = 0; d < 8; ++d) {
      float o = lg[10] * qp[d];                     // first term: attn[10] * q_; v terms added below
#pragma unroll
      for (int n = 0; n < 10; ++n) o += lg[n] * s_kv[(n * 16 + m) * LDQ + 64 + hh * 8 + d];
      s_ao[m * LDA + hh * 8 + d] = o * inv;
    }
  }
  __syncthreads();

  // ---- output projection: (16x64) @ (64x64); waves 0..3 (wave-uniform branch: EXEC stays full) ----
  if (wv < 4) {
    const int n0 = wv << 4;
    v2f bf[16];
#pragma unroll
    for (int kk = 0; kk < 16; ++kk) bf[kk] = b_frag(Wpr, 64, n0, lane, kk);
    v8f acc = {};
#pragma unroll
    for (int kk = 0; kk < 16; ++kk)
      acc = __builtin_amdgcn_wmma_f32_16x16x4_f32(
          false, a_frag(s_ao, lane, kk), false, bf[kk], (short)0, acc, false, false);
    const int c = n0 + (lane & 15);
    const float bias = bpr[c];
    float* __restrict__ ob = out + ((size_t)bi * 64 + c) * 4096 + h * 64 + w0;
#pragma unroll
    for (int g = 0; g < 8; ++g) {
      const int m = g + ((lane >> 4) << 3);
      ob[m] = acc[g] + bias;  // 8 consecutive floats per lane along w
    }
  }
}

extern "C" void kernel_launch(void* const* d_in, const int* in_sizes, int n_in,
                              void* d_out, int out_size, void* d_ws, size_t ws_size,
                              hipStream_t stream) {
  const float* x    = (const float*)d_in[0];
  const float* Wqk  = (const float*)d_in[1];
  const float* bqk  = (const float*)d_in[2];
  const float* Wkv  = (const float*)d_in[3];
  const float* bkv  = (const float*)d_in[4];
  const float* Wpr  = (const float*)d_in[5];
  const float* bpr  = (const float*)d_in[6];
  float* out = (float*)d_out;

  // B = b*H*W pixels = in_sizes[0] / C; 16 pixels per block.
  const int B = in_sizes[0] / 64;        // 32768
  const int blocks = B / 16;             // 2048
  pix_attn_fused_kernel<<<blocks, 256, 0, stream>>>(x, Wqk, bqk, Wkv, bkv, Wpr, bpr, out);
}